// EnergyScoreLoss_68805376082324
// MI455X (gfx1250) — compile-verified
//
#include <hip/hip_runtime.h>
#include <hip/hip_bf16.h>

typedef __attribute__((ext_vector_type(2))) float v2f;
typedef __attribute__((ext_vector_type(8))) float v8f;

#define NSAMP 50
#define NPAIRS (NSAMP * (NSAMP - 1) / 2)   // 1225
#define BLK 256

// Stage 1: one thread per (b,d) element.
//  - 53 coalesced streaming loads (mean/var/target + 50 noise slices)
//  - samples in registers, pairwise |x_i - x_j| fully unrolled
//    (abs folds into VALU source modifiers -> 2 VALU ops per pair)
//  - 4 rotating accumulators to break the serial add dependency chain
//  - deterministic LDS tree reduction -> one partial per block
__global__ void __launch_bounds__(BLK)
energy_score_stage1(const float* __restrict__ mean,
                    const float* __restrict__ variance,
                    const float* __restrict__ noise,
                    const float* __restrict__ target,
                    float* __restrict__ partials,
                    int E) {
    const int tid = threadIdx.x;
    const int e   = blockIdx.x * BLK + tid;

    float energy = 0.0f;
    if (e < E) {
        const float mu   = mean[e];
        const float var  = variance[e];
        const float tgt  = target[e];
        const float stdv = __builtin_sqrtf(var + 1e-6f);

        float x[NSAMP];
#pragma unroll
        for (int s = 0; s < NSAMP; ++s) {
            x[s] = noise[(size_t)s * (size_t)E + (size_t)e];
        }
#pragma unroll
        for (int s = 0; s < NSAMP; ++s) {
            x[s] = __builtin_fmaf(x[s], stdv, mu);
        }

        // first term: E_s |x_s - target|   (two accumulators for ILP)
        float f0 = 0.0f, f1 = 0.0f;
#pragma unroll
        for (int s = 0; s < NSAMP; s += 2) {
            f0 += __builtin_fabsf(x[s]     - tgt);
            f1 += __builtin_fabsf(x[s + 1] - tgt);
        }
        const float first = (f0 + f1) * (1.0f / (float)NSAMP);

        // second term: sum_{i<j} |x_i - x_j|  (sort identity, computed directly)
        float a0 = 0.0f, a1 = 0.0f, a2 = 0.0f, a3 = 0.0f;
#pragma unroll
        for (int i = 0; i < NSAMP; ++i) {
#pragma unroll
            for (int j = i + 1; j < NSAMP; ++j) {
                const float d = x[i] - x[j];
                const int k = j & 3;
                if (k == 0)      a0 += __builtin_fabsf(d);
                else if (k == 1) a1 += __builtin_fabsf(d);
                else if (k == 2) a2 += __builtin_fabsf(d);
                else             a3 += __builtin_fabsf(d);
            }
        }
        const float pair_sum = (a0 + a1) + (a2 + a3);
        const float second   = pair_sum * (1.0f / (float)NPAIRS);

        energy = first - 0.5f * second;   // BETA = 1.0
    }

    __shared__ float sdata[BLK];
    sdata[tid] = energy;
    __syncthreads();
#pragma unroll
    for (int s = BLK / 2; s > 0; s >>= 1) {
        if (tid < s) sdata[tid] += sdata[tid + s];
        __syncthreads();
    }
    if (tid == 0) partials[blockIdx.x] = sdata[0];
}

// Stage 2: single wave32 block. Matrix-pipe f32 reduction:
// A = all-ones 16x4, B = 4x16 chunk of partials, D = A*B + C  => every row of
// D holds the column sums of B, so sum(all D entries) == 16 * sum(B entries),
// independent of the exact B VGPR layout. Chain C across 64-value chunks.
//
// The WMMA loop covers only the 64-aligned prefix with UNCONDITIONAL b64
// loads (no EXEC masking anywhere near the WMMAs -> EXEC stays all-ones as
// required); the tail (empty for P=2048) is folded into lane 0's scalar sum.
__global__ void __launch_bounds__(32)
energy_score_stage2(const float* __restrict__ partials,
                    int P, float* __restrict__ out, float inv_count) {
    const int lane = threadIdx.x;   // 0..31

    v2f a; a.x = 1.0f; a.y = 1.0f;  // all-ones A (16x4 f32 = 2 VGPRs/lane)
    v8f c = {};

    const int iters = P / 64;                       // full 64-value chunks
    const v2f* __restrict__ p2 = (const v2f*)partials;
    for (int t = 0; t < iters; ++t) {
        v2f b = p2[t * 32 + lane];                  // one global_load_b64
        // D = A x B + C : v_wmma_f32_16x16x4_f32
        c = __builtin_amdgcn_wmma_f32_16x16x4_f32(
                /*neg_a=*/false, a, /*neg_b=*/false, b,
                /*c_mod=*/(short)0, c, /*reuse_a=*/false, /*reuse_b=*/false);
    }

    float s = ((c[0] + c[1]) + (c[2] + c[3])) + ((c[4] + c[5]) + (c[6] + c[7]));

    __shared__ float red[32];
    red[lane] = s;
    __syncthreads();
    if (lane == 0) {
        float tot = 0.0f;
        for (int i = 0; i < 32; ++i) tot += red[i];
        tot *= (1.0f / 16.0f);          // D rows duplicate the column sums
        for (int i = iters * 64; i < P; ++i) tot += partials[i];  // tail
        out[0] = tot * inv_count;
    }
}

extern "C" void kernel_launch(void* const* d_in, const int* in_sizes, int n_in,
                              void* d_out, int out_size, void* d_ws, size_t ws_size,
                              hipStream_t stream) {
    const float* mean     = (const float*)d_in[0];
    const float* variance = (const float*)d_in[1];
    const float* noise    = (const float*)d_in[2];
    const float* target   = (const float*)d_in[3];
    float* out = (float*)d_out;

    const int E = in_sizes[0];                 // B*D = 524288
    const int nblocks = (E + BLK - 1) / BLK;   // 2048 partials

    float* partials = (float*)d_ws;

    energy_score_stage1<<<nblocks, BLK, 0, stream>>>(mean, variance, noise, target,
                                                     partials, E);
    energy_score_stage2<<<1, 32, 0, stream>>>(partials, nblocks, out,
                                              1.0f / (float)E);
}